// MultiHeadAttention_11940009083380
// MI455X (gfx1250) — compile-verified
//
#include <hip/hip_runtime.h>

// MHA: B=2, S=2048, D=1024, H=16 (d=64), fp32 throughout via V_WMMA_F32_16X16X4_F32.
#define B_ 2
#define S_ 2048
#define D_ 1024
#define H_ 16
#define DH_ 64            // head dim
#define MASKVAL (-1.0e6f)

typedef __attribute__((ext_vector_type(2))) float v2f;
typedef __attribute__((ext_vector_type(8))) float v8f;

__device__ __forceinline__ v8f wmma4(v2f a, v2f b, v8f c) {
  // (neg_a, A, neg_b, B, c_mod, C, reuse_a, reuse_b)
  return __builtin_amdgcn_wmma_f32_16x16x4_f32(false, a, false, b, (short)0, c,
                                               false, false);
}

// ---------------------------------------------------------------------------
// Y = X @ W^T   computed as Y^T = W * X^T so that:
//   A-frag (W rows) : lane(n,h) loads W[col0+n][k0+2h..+1]  -> contiguous b64
//   B-frag (X rows) : lane(n,h) loads X[row0+n][k0+2h..+1]  -> contiguous b64
//   D tile [col=r+8h][row=n] -> row-major Y store = 2x float4 per tile per lane
// One wave computes a 64(rows) x 64(cols) block: 16 accumulator tiles.
// One-step fragment double-buffer (loads run one full 16-WMMA group ahead);
// unroll kept at 1: deeper unrolling was tried and either pushed allocation
// past 256 VGPRs (s_set_vgpr_msb in the hot loop) or forced scratch spills.
// This shape compiles to ~168 VGPRs, no spills, staggered loadcnt waits.
// X: (4096, 1024), W: (1024, 1024), Y: (4096, 1024)
// ---------------------------------------------------------------------------
__global__ __launch_bounds__(256, 4) void gemm_xwt_kernel(
    const float* __restrict__ X, const float* __restrict__ W,
    float* __restrict__ Y) {
  const int lane = threadIdx.x & 31;
  const int wave = (int)((blockIdx.x * blockDim.x + threadIdx.x) >> 5);
  const int n = lane & 15;
  const int h = lane >> 4;
  const int NT = D_ / 64;            // wave-tile columns = 16
  const int tm = wave / NT;
  const int tn = wave % NT;
  const int row0 = tm * 64;
  const int col0 = tn * 64;

  v8f acc[4][4] = {};                // [ci][ri] tiles of Y^T

  const float* Wb = W + (size_t)(col0 + n) * D_ + 2 * h;
  const float* Xb = X + (size_t)(row0 + n) * D_ + 2 * h;

  v2f aw[4], bx[4];
#pragma unroll
  for (int i = 0; i < 4; ++i) {
    aw[i] = *(const v2f*)(Wb + (size_t)i * 16 * D_);
    bx[i] = *(const v2f*)(Xb + (size_t)i * 16 * D_);
  }

#pragma unroll 1
  for (int k0 = 0; k0 < D_ - 4; k0 += 4) {
    v2f awn[4], bxn[4];
#pragma unroll
    for (int i = 0; i < 4; ++i) {     // prefetch next k-step (8 x b64)
      awn[i] = *(const v2f*)(Wb + (size_t)i * 16 * D_ + k0 + 4);
      bxn[i] = *(const v2f*)(Xb + (size_t)i * 16 * D_ + k0 + 4);
    }
#pragma unroll
    for (int ci = 0; ci < 4; ++ci)
#pragma unroll
      for (int ri = 0; ri < 4; ++ri)
        acc[ci][ri] = wmma4(aw[ci], bx[ri], acc[ci][ri]);
#pragma unroll
    for (int i = 0; i < 4; ++i) {
      aw[i] = awn[i];
      bx[i] = bxn[i];
    }
  }
#pragma unroll
  for (int ci = 0; ci < 4; ++ci)      // last k-step
#pragma unroll
    for (int ri = 0; ri < 4; ++ri)
      acc[ci][ri] = wmma4(aw[ci], bx[ri], acc[ci][ri]);

#pragma unroll
  for (int ri = 0; ri < 4; ++ri) {
    float* yrow = Y + (size_t)(row0 + ri * 16 + n) * D_ + col0;
#pragma unroll
    for (int ci = 0; ci < 4; ++ci) {
      float* dst = yrow + ci * 16 + 8 * h;   // D tile element [m=r+8h][n]
      v8f t = acc[ci][ri];
      *(float4*)dst = make_float4(t[0], t[1], t[2], t[3]);
      *(float4*)(dst + 4) = make_float4(t[4], t[5], t[6], t[7]);
    }
  }
}

// ---------------------------------------------------------------------------
// Flash attention, one wave per (head-batch, 16-query block).
// Q/K/V are the projection buffers (B*S, D); head hd is the column slice
// [hd*64, hd*64+64).  Scores computed transposed: S^T = K * Q^T  (keys in
// accumulator rows, queries in lanes) so the softmax reduction per query is
// 8 regs + one shfl_xor(16).  O^T = V^T * P^T; P^T B-fragments are built from
// the score registers with 8 shfl_xor(16) + per-lane selects (no LDS).
// ---------------------------------------------------------------------------
__global__ __launch_bounds__(256) void attn_kernel(
    const float* __restrict__ Q, const float* __restrict__ K,
    const float* __restrict__ V, const int* __restrict__ valid_len,
    float* __restrict__ O) {
  const int lane = threadIdx.x & 31;
  const int wave = (int)((blockIdx.x * blockDim.x + threadIdx.x) >> 5);
  const int n = lane & 15;
  const int h = lane >> 4;

  const int QB = S_ / 16;            // 128 query blocks per head
  const int qb = wave % QB;
  const int bh = wave / QB;          // 0..B*H-1
  const int b = bh / H_;
  const int hd = bh % H_;
  const int vl = valid_len[b];
  const int q0 = qb * 16;

  const size_t base = (size_t)b * S_ * D_ + (size_t)hd * DH_;
  const float* Qh = Q + base;        // [s][dd] at s*D_ + dd
  const float* Kh = K + base;
  const float* Vh = V + base;

  // Loop-invariant Q^T B-fragments: b.x = Q[q0+n][4j+2h], b.y = +1
  v2f qf[16];
  const float* qrow = Qh + (size_t)(q0 + n) * D_ + 2 * h;
#pragma unroll
  for (int j = 0; j < 16; ++j) qf[j] = *(const v2f*)(qrow + 4 * j);

  v8f o[4] = {};                     // O^T tiles: d block i, [d=i*16+r+8h][q=n]
  float mval = -3.0e38f;
  float lsum = 0.0f;

  const int nkb = (vl + 15) >> 4;    // key blocks with any valid key
  for (int kb = 0; kb < nkb; ++kb) {
    const int key0 = kb * 16;

    // S^T tile (16 keys x 16 queries): A = K rows, B = Q^T fragments
    v8f s = {};
    const float* krow = Kh + (size_t)(key0 + n) * D_ + 2 * h;
#pragma unroll
    for (int j = 0; j < 16; ++j) {
      v2f a = *(const v2f*)(krow + 4 * j);
      s = wmma4(a, qf[j], s);
    }

    // Scale, mask, per-query online softmax (query = lane n, keys = r+8h)
    float p[8];
    float tmax = -3.0e38f;
#pragma unroll
    for (int r = 0; r < 8; ++r) {
      float sv = s[r] * 0.125f;      // 1/sqrt(64)
      int key = key0 + r + 8 * h;
      sv = (key < vl) ? sv : MASKVAL;
      p[r] = sv;
      tmax = fmaxf(tmax, sv);
    }
    tmax = fmaxf(tmax, __shfl_xor(tmax, 16, 32));
    float mnew = fmaxf(mval, tmax);
    float alpha = __expf(mval - mnew);
    float rsum = 0.0f;
#pragma unroll
    for (int r = 0; r < 8; ++r) {
      p[r] = __expf(p[r] - mnew);
      rsum += p[r];
    }
    rsum += __shfl_xor(rsum, 16, 32);
    lsum = lsum * alpha + rsum;
    mval = mnew;
#pragma unroll
    for (int i = 0; i < 4; ++i)
#pragma unroll
      for (int r = 0; r < 8; ++r) o[i][r] *= alpha;

    // Build P^T B-fragments.  P^T[k][q] lives at lane q+16*(k>>3), reg k&7.
    // B-frag kk: b.x = P^T[kk+2h][n], b.y = P^T[kk+2h+1][n].
    float sh[8];
#pragma unroll
    for (int r = 0; r < 8; ++r) sh[r] = __shfl_xor(p[r], 16, 32);
    v2f pb[4];
    pb[0][0] = h ? sh[2] : p[0];  pb[0][1] = h ? sh[3] : p[1];   // kk=0
    pb[1][0] = h ? sh[6] : p[4];  pb[1][1] = h ? sh[7] : p[5];   // kk=4
    pb[2][0] = h ? p[2] : sh[0];  pb[2][1] = h ? p[3] : sh[1];   // kk=8
    pb[3][0] = h ? p[6] : sh[4];  pb[3][1] = h ? p[7] : sh[5];   // kk=12

    // O^T += V^T * P^T : A-frag a.x = V[key0+4j2+2h][i*16+n]
    const float* vbase = Vh + (size_t)key0 * D_;
#pragma unroll
    for (int j2 = 0; j2 < 4; ++j2) {
      const int kr = 4 * j2 + 2 * h;
#pragma unroll
      for (int i = 0; i < 4; ++i) {
        v2f a;
        a[0] = vbase[(size_t)kr * D_ + i * 16 + n];
        a[1] = vbase[(size_t)(kr + 1) * D_ + i * 16 + n];
        o[i] = wmma4(a, pb[j2], o[i]);
      }
    }
  }

  // Store normalized output to (B, S, D) with heads concatenated.
  const float inv = 1.0f / lsum;     // vl >= 1 guarantees lsum > 0
  float* orow = O + ((size_t)b * S_ + q0 + n) * D_ + (size_t)hd * DH_ + 8 * h;
#pragma unroll
  for (int i = 0; i < 4; ++i) {
    v8f t = o[i];
    *(float4*)(orow + i * 16) =
        make_float4(t[0] * inv, t[1] * inv, t[2] * inv, t[3] * inv);
    *(float4*)(orow + i * 16 + 4) =
        make_float4(t[4] * inv, t[5] * inv, t[6] * inv, t[7] * inv);
  }
}

// ---------------------------------------------------------------------------
extern "C" void kernel_launch(void* const* d_in, const int* in_sizes, int n_in,
                              void* d_out, int out_size, void* d_ws,
                              size_t ws_size, hipStream_t stream) {
  const float* query = (const float*)d_in[0];
  const float* key   = (const float*)d_in[1];
  const float* value = (const float*)d_in[2];
  const int* vlen    = (const int*)d_in[3];
  const float* Wq    = (const float*)d_in[4];
  const float* Wk    = (const float*)d_in[5];
  const float* Wv    = (const float*)d_in[6];
  const float* Wo    = (const float*)d_in[7];
  float* out = (float*)d_out;

  float* ws = (float*)d_ws;
  const size_t stride = (size_t)B_ * S_ * D_;  // 4M floats per buffer
  float* Qp = ws;
  float* Kp = ws + stride;
  float* Vp = ws + 2 * stride;
  float* Ao = ws + 3 * stride;

  const int M = B_ * S_;                         // 4096
  const int gemm_waves = (M / 64) * (D_ / 64);   // 1024 waves
  const int gemm_blocks = gemm_waves / 8;        // 128 blocks x 256 threads
  const int attn_waves = B_ * H_ * (S_ / 16);    // 4096 waves
  const int attn_blocks = attn_waves / 8;        // 512 blocks

  gemm_xwt_kernel<<<gemm_blocks, 256, 0, stream>>>(query, Wq, Qp);
  gemm_xwt_kernel<<<gemm_blocks, 256, 0, stream>>>(key,   Wk, Kp);
  gemm_xwt_kernel<<<gemm_blocks, 256, 0, stream>>>(value, Wv, Vp);
  attn_kernel<<<attn_blocks, 256, 0, stream>>>(Qp, Kp, Vp, vlen, Ao);
  gemm_xwt_kernel<<<gemm_blocks, 256, 0, stream>>>(Ao, Wo, out);
}